// MultiHeadSelfAttention_88862873354447
// MI455X (gfx1250) — compile-verified
//
#include <hip/hip_runtime.h>
#include <hip/hip_bf16.h>
#include <math.h>

// Problem constants (from reference setup_inputs)
#define BB   4
#define NN   2048
#define EE   768
#define HH   16
#define N3   48          // 3*H
#define MM   (BB*NN)     // 8192 rows

typedef __attribute__((ext_vector_type(2))) float v2f;
typedef __attribute__((ext_vector_type(8))) float v8f;

// ---------------------------------------------------------------------------
// gfx1250 async copy: global -> LDS, 4 bytes per lane, tracked by ASYNCcnt.
// (inline asm per CDNA5_HIP.md: portable across ROCm 7.2 / amdgpu-toolchain)
// ---------------------------------------------------------------------------
__device__ __forceinline__ void async_g2lds_b32(const float* g, float* l) {
    // flat AS3->generic pointers carry the LDS byte offset in the low 32 bits
    unsigned           lds = (unsigned)(uintptr_t)l;
    unsigned long long ga  = (unsigned long long)(uintptr_t)g;
    asm volatile("global_load_async_to_lds_b32 %0, %1, off"
                 :: "v"(lds), "v"(ga) : "memory");
}
__device__ __forceinline__ void wait_asynccnt0() {
    asm volatile("s_wait_asynccnt 0x0" ::: "memory");
}

// ---------------------------------------------------------------------------
// Kernel 1: qkv[m, 0:48] = x[m, 0:768] . w_qkv[0:48, 0:768]^T   (f32 WMMA)
// One wave computes one 16x16 tile of the [8192 x 48] result.
// A (16x4 f32): lane L holds row L%16, K = 2*(L/16)+{0,1}  -> float2 load
// B (4x16 f32): lane L holds col L%16, K = 2*(L/16)+{0,1}  -> float2 load
// ---------------------------------------------------------------------------
__global__ void __launch_bounds__(256)
qkv_gemm_wmma(const float* __restrict__ x, const float* __restrict__ w,
              float* __restrict__ qkv) {
    const int wave = (blockIdx.x * blockDim.x + threadIdx.x) >> 5;
    const int lane = threadIdx.x & 31;
    const int ntn  = N3 >> 4;              // 3 tiles along the 48-dim
    const int tm   = wave / ntn;
    const int tn   = wave % ntn;
    if (tm * 16 >= MM) return;             // wave-uniform: EXEC stays all-1s

    const int half = lane >> 4;            // 0: K pair {0,1}, 1: K pair {2,3}
    const int arow = tm * 16 + (lane & 15);
    const int bcol = tn * 16 + (lane & 15);
    const float* __restrict__ ap = x + (size_t)arow * EE + 2 * half;
    const float* __restrict__ bp = w + (size_t)bcol * EE + 2 * half;

    v8f c = {};
    for (int kk = 0; kk < EE; kk += 64) {
        // gfx1250 global_prefetch_b8 one chunk ahead (high temporal locality)
        __builtin_prefetch(ap + kk + 64, 0, 3);
        __builtin_prefetch(bp + kk + 64, 0, 3);
#pragma unroll
        for (int k = 0; k < 64; k += 4) {
            v2f a = *(const v2f*)(ap + kk + k);
            v2f b = *(const v2f*)(bp + kk + k);
            // D = A(16x4) x B(4x16) + C   -> v_wmma_f32_16x16x4_f32
            c = __builtin_amdgcn_wmma_f32_16x16x4_f32(
                    false, a, false, b, (short)0, c, false, false);
        }
    }

    // C/D layout: VGPR r, lanes 0-15 -> M=r, lanes 16-31 -> M=r+8; N = lane%16
    const int rbase = tm * 16 + 8 * half;
#pragma unroll
    for (int r = 0; r < 8; ++r)
        qkv[(size_t)(rbase + r) * N3 + bcol] = c[r];
}

// ---------------------------------------------------------------------------
// Kernel 2: head_dim-1 attention with masked diagonal.
// dots[n,m] = q_n * k_m (rank-1); diagonal excluded (mask = -FLT_MAX -> w=0).
// k,v for one (b,h) are 2048 floats each -> staged in LDS via
// global_load_async_to_lds_b32 (ASYNCcnt) then s_wait_asynccnt + barrier.
// One wave per row n; two-pass softmax in base 2:
//   pass 1: row max over m!=n (pure v_max_f32 scan),
//   pass 2: one v_exp_f32 + two FMAs per element (half the TRANS work of
//           an online-softmax formulation).
// ---------------------------------------------------------------------------
__global__ void __launch_bounds__(256)
attn_head1(const float* __restrict__ qkv, float* __restrict__ ao) {
    __shared__ float ks[NN];
    __shared__ float vs[NN];

    const int bh = blockIdx.x;             // 0..B*H-1
    const int b  = bh / HH;
    const int h  = bh % HH;
    const float* __restrict__ base = qkv + (size_t)b * NN * N3;

    for (int i = threadIdx.x; i < NN; i += blockDim.x) {
        async_g2lds_b32(&base[i * N3 + HH + h],     &ks[i]);   // k slice
        async_g2lds_b32(&base[i * N3 + 2 * HH + h], &vs[i]);   // v slice
    }
    wait_asynccnt0();
    __syncthreads();

    const int wave = threadIdx.x >> 5;
    const int lane = threadIdx.x & 31;
    const int rows = NN / gridDim.y;         // rows per block
    const int row0 = blockIdx.y * rows;
    const float LOG2E = 1.4426950408889634f;

    for (int n = row0 + wave; n < row0 + rows; n += 8) {
        const float q2 = base[n * N3 + h] * LOG2E;   // softmax in base 2

        // ---- pass 1: mx = max_{m != n} q2*k[m]  (branchless diagonal mask)
        float mx = -INFINITY;
        for (int m = lane; m < NN; m += 32) {
            float s = q2 * ks[m];
            mx = (m != n) ? fmaxf(mx, s) : mx;       // v_cndmask
        }
#pragma unroll
        for (int off = 16; off > 0; off >>= 1)
            mx = fmaxf(mx, __shfl_xor(mx, off, 32));

        // ---- pass 2: single exp2 per element, fused accumulate
        float l = 0.f, acc = 0.f;
        for (int m = lane; m < NN; m += 32) {
            float p = __builtin_amdgcn_exp2f(q2 * ks[m] - mx);   // v_exp_f32
            p   = (m != n) ? p : 0.f;                // masked diagonal
            l  += p;
            acc = fmaf(p, vs[m], acc);
        }
#pragma unroll
        for (int off = 16; off > 0; off >>= 1) {
            l   += __shfl_xor(l,   off, 32);
            acc += __shfl_xor(acc, off, 32);
        }
        if (lane == 0)
            ao[(size_t)(b * NN + n) * HH + h] = acc / l;
    }
}

// ---------------------------------------------------------------------------
// Kernel 3: out[m, 0:768] = ao[m, 0:16] . w_out[0:768, 0:16]^T + b_out
// K = 16 -> exactly four v_wmma_f32_16x16x4_f32 per 16x16 tile.
// ---------------------------------------------------------------------------
__global__ void __launch_bounds__(256)
out_gemm_wmma(const float* __restrict__ ao, const float* __restrict__ w_out,
              const float* __restrict__ b_out, float* __restrict__ out) {
    const int wave = (blockIdx.x * blockDim.x + threadIdx.x) >> 5;
    const int lane = threadIdx.x & 31;
    const int ntn  = EE >> 4;              // 48 tiles along E
    const int tm   = wave / ntn;
    const int tn   = wave % ntn;
    if (tm * 16 >= MM) return;

    const int half = lane >> 4;
    const int arow = tm * 16 + (lane & 15);
    const int ecol = tn * 16 + (lane & 15);
    const float* __restrict__ ap = ao    + (size_t)arow * HH + 2 * half;
    const float* __restrict__ bp = w_out + (size_t)ecol * HH + 2 * half;

    v8f c = {};
#pragma unroll
    for (int k = 0; k < HH; k += 4) {
        v2f a = *(const v2f*)(ap + k);
        v2f b = *(const v2f*)(bp + k);
        c = __builtin_amdgcn_wmma_f32_16x16x4_f32(
                false, a, false, b, (short)0, c, false, false);
    }

    const float bias  = b_out[ecol];
    const int   rbase = tm * 16 + 8 * half;
#pragma unroll
    for (int r = 0; r < 8; ++r)
        out[(size_t)(rbase + r) * EE + ecol] = c[r] + bias;
}

// ---------------------------------------------------------------------------
extern "C" void kernel_launch(void* const* d_in, const int* in_sizes, int n_in,
                              void* d_out, int out_size, void* d_ws, size_t ws_size,
                              hipStream_t stream) {
    const float* x     = (const float*)d_in[0];   // [4,2048,768]
    const float* w_qkv = (const float*)d_in[1];   // [48,768]
    const float* w_out = (const float*)d_in[2];   // [768,16]
    const float* b_out = (const float*)d_in[3];   // [768]
    float*       out   = (float*)d_out;           // [4,2048,768]

    float* qkv = (float*)d_ws;                    // [8192,48]  1.5 MB
    float* ao  = qkv + (size_t)MM * N3;           // [8192,16]  0.5 MB

    // Phase 1: QKV projection. (8192/16)*3 = 1536 waves, 8 waves/block.
    qkv_gemm_wmma<<<1536 / 8, 256, 0, stream>>>(x, w_qkv, qkv);

    // Phase 2: attention. 64 (b,h) pairs x 16 row-chunks of 128 rows.
    attn_head1<<<dim3(BB * HH, 16), 256, 0, stream>>>(qkv, ao);

    // Phase 3: output projection + bias. 512*48 = 24576 waves.
    out_gemm_wmma<<<24576 / 8, 256, 0, stream>>>(ao, w_out, b_out, out);
}